// PostProcessing_9766755631845
// MI455X (gfx1250) — compile-verified
//
#include <hip/hip_runtime.h>
#include <math.h>

#define NCLS 10
#define DROW 22
#define MTOP 4096
#define CAND_MAX 8192
#define TOPK 100
#define ROWS_TILE 256
#define TILES_PB 16
#define ROWS_BLOCK (ROWS_TILE * TILES_PB)
#define NBIN 4096

typedef unsigned int v4u __attribute__((ext_vector_type(4)));
typedef int v8i __attribute__((ext_vector_type(8)));
typedef int v4i __attribute__((ext_vector_type(4)));

// monotone float -> uint key (ascending)
__device__ __forceinline__ unsigned okey(float f){
  unsigned u = __float_as_uint(f);
  return (u & 0x80000000u) ? ~u : (u | 0x80000000u);
}

// TDM issue (no wait): DMA a 2D tile (rows x tileElems f32, row stride
// strideElems) from global into LDS at byte offset ldsOff, compacted.
// D# layout per CDNA5 ISA ch.8 (groups 0/1; groups 2/3 unused -> 2D).
__device__ __forceinline__ void tdm_issue(unsigned ldsOff, const float* gsrc,
                                          int rows, int tensorD0, int tileElems,
                                          int strideElems){
  unsigned long long ga = (unsigned long long)(size_t)gsrc;
  v4u g0; v8i g1; v4i z4; v8i z8;
  g0[0] = 1u;                                            // count=1 (valid), no gather
  g0[1] = ldsOff;                                        // lds_addr (bytes)
  g0[2] = (unsigned)ga;                                  // global_addr lo32
  g0[3] = (unsigned)((ga >> 32) & 0x1FFFFFFull) | 0x80000000u; // addr hi + type=2
  g1[0] = (int)(2u << 16);                               // data_size = 4B
  g1[1] = (int)(((unsigned)tensorD0 & 0xFFFFu) << 16);   // tensor_dim0 lo16
  g1[2] = (int)((((unsigned)tensorD0 >> 16) & 0xFFFFu) |
                (((unsigned)rows & 0xFFFFu) << 16));     // dim0 hi | tensor_dim1 lo
  g1[3] = (int)((((unsigned)rows >> 16) & 0xFFFFu) |
                (((unsigned)tileElems & 0xFFFFu) << 16));// dim1 hi | tile_dim0
  g1[4] = (int)((unsigned)rows & 0xFFFFu);               // tile_dim1 (tile_dim2=0)
  g1[5] = strideElems;                                   // tensor_dim0_stride lo32
  g1[6] = 0; g1[7] = 0;
  z4[0] = z4[1] = z4[2] = z4[3] = 0;
  z8[0] = z8[1] = z8[2] = z8[3] = z8[4] = z8[5] = z8[6] = z8[7] = 0;
  __builtin_amdgcn_tensor_load_to_lds(g0, g1, z4, z4, z8, 0);
}

// Pass 1: stream batch-0 rows via double-buffered TDM; per-row score key +
// LDS radix histogram, amortized over ROWS_BLOCK rows per block.
__global__ void __launch_bounds__(256) k_decode_hist(const float* __restrict__ y,
    unsigned* __restrict__ keys, unsigned* __restrict__ hist, int n){
  __shared__ float tile[2][ROWS_TILE * NCLS];
  __shared__ unsigned lh[NBIN];
  int tid = threadIdx.x;
  for (int b = tid; b < NBIN; b += 256) lh[b] = 0u;
  int base = blockIdx.x * ROWS_BLOCK;

  int rows0 = n - base; if (rows0 > ROWS_TILE) rows0 = ROWS_TILE; if (rows0 < 0) rows0 = 0;
  if (tid < 32 && rows0 > 0)
    tdm_issue((unsigned)(size_t)(void*)&tile[0][0], y + (size_t)base * DROW,
              rows0, DROW, NCLS, DROW);

  for (int t = 0; t < TILES_PB; ++t){
    int row0 = base + t * ROWS_TILE;
    int rows = n - row0; if (rows > ROWS_TILE) rows = ROWS_TILE;
    if (rows <= 0) break;                       // uniform
    int nrow0 = row0 + ROWS_TILE;
    int nrows = n - nrow0; if (nrows > ROWS_TILE) nrows = ROWS_TILE; if (nrows < 0) nrows = 0;
    bool more = (t + 1 < TILES_PB) && (nrows > 0);
    if (tid < 32){
      if (more){                                 // prefetch next tile, then wait
        tdm_issue((unsigned)(size_t)(void*)&tile[(t + 1) & 1][0],
                  y + (size_t)nrow0 * DROW, nrows, DROW, NCLS, DROW);
        __builtin_amdgcn_s_wait_tensorcnt(1);    // in-order: current tile done
      } else {
        __builtin_amdgcn_s_wait_tensorcnt(0);
      }
    }
    __syncthreads();
    if (tid < rows){
      const float* tp = &tile[t & 1][tid * NCLS];
      float best = tp[0]; int bi = 0;
      #pragma unroll
      for (int c = 1; c < NCLS; ++c){ float v = tp[c]; if (v > best){ best = v; bi = c; } }
      bool valid = (bi != 0) && (best != 0.5f);
      float score = valid ? best : -1e30f;
      unsigned kk = okey(score);
      keys[row0 + tid] = kk;
      atomicAdd(&lh[kk >> 20], 1u);
    }
    __syncthreads();                             // buffer reusable at t+2
  }
  for (int b = tid; b < NBIN; b += 256){ unsigned c = lh[b]; if (c) atomicAdd(&hist[b], c); }
}

// Pass 2: find key-prefix threshold holding >= MTOP entries
__global__ void k_thresh(const unsigned* __restrict__ hist, unsigned* __restrict__ ctl){
  if (threadIdx.x == 0){
    unsigned cum = 0, thr = 0;
    for (int b = NBIN - 1; b >= 0; --b){
      cum += hist[b];
      if (cum >= (unsigned)MTOP){ thr = ((unsigned)b) << 20; break; }
    }
    ctl[0] = thr;
  }
}

// Pass 3: compact candidates (key, ~idx) above threshold
__global__ void __launch_bounds__(256) k_compact(const unsigned* __restrict__ keys,
    unsigned* __restrict__ ctl, unsigned long long* __restrict__ cand, int n){
  unsigned thr = ctl[0];
  int stride = gridDim.x * blockDim.x;
  for (int i = blockIdx.x * blockDim.x + threadIdx.x; i < n; i += stride){
    unsigned k = keys[i];
    if (k >= thr){
      unsigned p = atomicAdd(&ctl[1], 1u);
      if (p < (unsigned)CAND_MAX)
        cand[p] = ((unsigned long long)k << 32) | (unsigned)(~(unsigned)i);
    }
  }
}

// Pass 4: bitonic sort (desc key, asc index) in LDS; gather + re-decode top 4096
__global__ void __launch_bounds__(1024) k_sort_gather(const float* __restrict__ y,
    const unsigned* __restrict__ ctl, const unsigned long long* __restrict__ cand,
    float4* __restrict__ bx, float* __restrict__ cls, float* __restrict__ sc,
    unsigned* __restrict__ va){
  extern __shared__ unsigned long long s4[];
  int tid = threadIdx.x;
  unsigned nc = ctl[1]; if (nc > (unsigned)CAND_MAX) nc = CAND_MAX;
  for (int i = tid; i < CAND_MAX; i += 1024) s4[i] = (i < (int)nc) ? cand[i] : 0ull;
  __syncthreads();
  for (unsigned k = 2; k <= (unsigned)CAND_MAX; k <<= 1){
    for (unsigned j = k >> 1; j > 0; j >>= 1){
      for (unsigned i = tid; i < (unsigned)CAND_MAX; i += 1024){
        unsigned ixj = i ^ j;
        if (ixj > i){
          unsigned long long a = s4[i], b = s4[ixj];
          bool desc = ((i & k) == 0);
          if (desc ? (a < b) : (a > b)){ s4[i] = b; s4[ixj] = a; }
        }
      }
      __syncthreads();
    }
  }
  for (int t = tid; t < MTOP; t += 1024){
    float4 bb = make_float4(0.f, 0.f, 0.f, 0.f);
    float cc = 0.f, ss = -1e30f; unsigned vv = 0u;
    if (t < (int)nc){
      unsigned long long e = s4[t];
      unsigned idx = ~(unsigned)(e & 0xFFFFFFFFull);
      const float* rp = y + (size_t)idx * DROW;
      float best = rp[0]; int bi = 0;
      #pragma unroll
      for (int c = 1; c < NCLS; ++c){ float v = rp[c]; if (v > best){ best = v; bi = c; } }
      float y10=rp[10], y11=rp[11], y12=rp[12], y13=rp[13], y14=rp[14], y15=rp[15],
            y16=rp[16], y17=rp[17], y18=rp[18], y19=rp[19], y20=rp[20], y21=rp[21];
      float w1 = expf(y12 * y20), h1 = expf(y13 * y21);
      float w = w1 + w1 * y16, h = h1 + h1 * y17;
      float cx = y14 + y10 * y18 * y16;
      float cy = y15 + y11 * y19 * y17;
      float xmn = fmaxf(cx - 0.5f * w, 0.f), xmx = fminf(cx + 0.5f * w, 511.f);
      float ymn = fmaxf(cy - 0.5f * h, 0.f), ymx = fminf(cy + 0.5f * h, 511.f);
      bb = make_float4(ymn, xmn, ymx, xmx);
      cc = (float)bi; ss = best; vv = (bi != 0 && best != 0.5f) ? 1u : 0u;
    }
    bx[t] = bb; cls[t] = cc; sc[t] = ss; va[t] = vv;
  }
}

// Pass 5: sequential NMS (single WG, early exit), class sort, output (B,100,6)
__global__ void __launch_bounds__(1024) k_nms_out(const float4* __restrict__ bxWS,
    const float* __restrict__ clsWS, const float* __restrict__ scWS,
    const unsigned* __restrict__ vaWS, float* __restrict__ out, int B){
  extern __shared__ unsigned char sm[];
  float4* bx = (float4*)sm;                       // 65536
  float*  ar = (float*)(sm + 65536);              // 16384
  unsigned char* sup = sm + 81920;                // 4096
  unsigned char* val = sm + 86016;                // 4096
  int*   sel = (int*)(sm + 90112);                // 100*4
  float* scl = (float*)(sm + 90624);              // 100*4
  int*   ord = (int*)(sm + 91136);                // 100*4
  int*   cnt = (int*)(sm + 91648);                // 4
  int tid = threadIdx.x;
  for (int j = tid; j < MTOP; j += 1024){
    float4 b = bxWS[j];
    bx[j] = b;
    ar[j] = fmaxf(b.z - b.x, 0.f) * fmaxf(b.w - b.y, 0.f);
    sup[j] = 0; val[j] = (unsigned char)(vaWS[j] ? 1 : 0);
  }
  if (tid == 0) *cnt = 0;
  __syncthreads();
  for (int i = 0; i < MTOP; ++i){
    int c = *cnt;                 // uniform: shared, barrier-protected
    if (c >= TOPK) break;
    bool can = (!sup[i]) && val[i];
    if (can){
      __syncthreads();            // all reads of sup[i]/cnt done before writes
      if (tid == 0){ sel[c] = i; *cnt = c + 1; }
      float4 bi = bx[i]; float ai = ar[i];
      for (int j = tid; j < MTOP; j += 1024){
        float4 bj = bx[j];
        float iy = fmaxf(fminf(bi.z, bj.z) - fmaxf(bi.x, bj.x), 0.f);
        float ix = fmaxf(fminf(bi.w, bj.w) - fmaxf(bi.y, bj.y), 0.f);
        float inter = iy * ix;
        float uni = ai + ar[j] - inter;
        if (inter > 0.25f * fmaxf(uni, 1e-9f)) sup[j] = 1;  // iou > IOU_THR
      }
      __syncthreads();
    }
  }
  __syncthreads();
  int c2 = *cnt;
  for (int k = tid; k < c2; k += 1024) scl[k] = clsWS[sel[k]];
  __syncthreads();
  if (tid == 0){                  // stable counting sort by class (1..9)
    int m = 0;
    for (int cv = 0; cv < 16 && m < c2; ++cv)
      for (int k = 0; k < c2; ++k)
        if ((int)scl[k] == cv) ord[m++] = k;
  }
  __syncthreads();
  int total = B * TOPK * 6;
  for (int t = tid; t < total; t += 1024){
    int rem = t % (TOPK * 6); int k = rem / 6; int c6 = rem % 6;
    float v = 0.f;
    if (k < c2){
      int ci = sel[ord[k]];
      if (c6 == 0) v = clsWS[ci];
      else if (c6 == 1) v = scWS[ci];
      else {
        float4 b = bx[ci];
        if (c6 == 2)      v = b.x * (1.f / 512.f);
        else if (c6 == 3) v = b.y * (1.f / 512.f);
        else if (c6 == 4) v = 1.f - b.z * (1.f / 512.f);
        else              v = 1.f - b.w * (1.f / 512.f);
      }
    }
    out[t] = v;
  }
}

extern "C" void kernel_launch(void* const* d_in, const int* in_sizes, int n_in,
                              void* d_out, int out_size, void* d_ws, size_t ws_size,
                              hipStream_t stream){
  const float* y = (const float*)d_in[0];
  int B = out_size / 600; if (B < 1) B = 1;
  int n = in_sizes[0] / (B * DROW);               // rows in batch 0
  char* ws = (char*)d_ws;
  size_t offH    = (((size_t)n * 4) + 255) & ~(size_t)255;
  size_t offC    = offH + (size_t)NBIN * 4;
  size_t offCand = offC + 256;
  size_t offBx   = offCand + (size_t)CAND_MAX * 8;
  size_t offCls  = offBx + (size_t)MTOP * 16;
  size_t offSc   = offCls + (size_t)MTOP * 4;
  size_t offVa   = offSc + (size_t)MTOP * 4;
  unsigned* keys = (unsigned*)ws;
  unsigned* hist = (unsigned*)(ws + offH);
  unsigned* ctl  = (unsigned*)(ws + offC);
  unsigned long long* cand = (unsigned long long*)(ws + offCand);
  float4* bxWS = (float4*)(ws + offBx);
  float* clsWS = (float*)(ws + offCls);
  float* scWS  = (float*)(ws + offSc);
  unsigned* vaWS = (unsigned*)(ws + offVa);

  (void)hipMemsetAsync(hist, 0, NBIN * 4 + 256, stream); // zero hist + ctl each call
  int nBlocks = (n + ROWS_BLOCK - 1) / ROWS_BLOCK;
  k_decode_hist<<<nBlocks, 256, 0, stream>>>(y, keys, hist, n);
  k_thresh<<<1, 32, 0, stream>>>(hist, ctl);
  k_compact<<<1024, 256, 0, stream>>>(keys, ctl, cand, n);
  k_sort_gather<<<1, 1024, (size_t)CAND_MAX * 8, stream>>>(y, ctl, cand, bxWS, clsWS, scWS, vaWS);
  k_nms_out<<<1, 1024, 92 * 1024, stream>>>(bxWS, clsWS, scWS, vaWS, (float*)d_out, B);
}